// GCNClassifier_for_dml_74509092651322
// MI455X (gfx1250) — compile-verified
//
#include <hip/hip_runtime.h>
#include <hip/hip_bf16.h>

#define N_NODES   100000
#define N_EDGES   1600000
#define N_GRAPHS  128
#define HIDDEN    128
#define N_CLASSES 10

typedef __attribute__((ext_vector_type(2))) float v2f;
typedef __attribute__((ext_vector_type(8))) float v8f;

// ---------- degree accumulation ----------
__global__ void deg_kernel(const int* __restrict__ src, const int* __restrict__ dst,
                           float* __restrict__ outdeg, float* __restrict__ indeg) {
    int e = blockIdx.x * blockDim.x + threadIdx.x;
    if (e < N_EDGES) {
        atomicAdd(&outdeg[src[e]], 1.0f);
        atomicAdd(&indeg[dst[e]], 1.0f);
    }
}

// ---------- per-node norms ----------
__global__ void norm_kernel(const float* __restrict__ outdeg, const float* __restrict__ indeg,
                            float* __restrict__ cs, float* __restrict__ cd) {
    int n = blockIdx.x * blockDim.x + threadIdx.x;
    if (n < N_NODES) {
        cs[n] = rsqrtf(fmaxf(outdeg[n], 1.0f));
        cd[n] = rsqrtf(fmaxf(indeg[n], 1.0f));
    }
}

// ---------- layer-1 scalar scatter: agg1[dst] += indeg[src]*cs[src] ----------
__global__ void scatter1_kernel(const int* __restrict__ src, const int* __restrict__ dst,
                                const float* __restrict__ indeg, const float* __restrict__ cs,
                                float* __restrict__ agg1) {
    int e = blockIdx.x * blockDim.x + threadIdx.x;
    if (e < N_EDGES) {
        int s = src[e];
        atomicAdd(&agg1[dst[e]], indeg[s] * cs[s]);
    }
}

// ---------- layer-1 rank-1 dense: h[m][j] = relu(agg1[m]*cd[m]*W1[j] + b1[j]) ----------
__global__ void layer1_kernel(const float* __restrict__ agg1, const float* __restrict__ cd,
                              const float* __restrict__ W1, const float* __restrict__ b1,
                              float* __restrict__ h) {
    int i = blockIdx.x * blockDim.x + threadIdx.x;   // over N_NODES*HIDDEN
    if (i < N_NODES * HIDDEN) {
        int m = i >> 7;
        int j = i & (HIDDEN - 1);
        float x = agg1[m] * cd[m];
        h[i] = fmaxf(fmaf(x, W1[j], b1[j]), 0.0f);
    }
}

// ---------- edge scatter (feature layers): one wave32 per edge ----------
__global__ void scatter_kernel(const int* __restrict__ src, const int* __restrict__ dst,
                               const float* __restrict__ cs, const float* __restrict__ h,
                               float* __restrict__ agg) {
    int gid  = blockIdx.x * blockDim.x + threadIdx.x;
    int e    = gid >> 5;
    int lane = gid & 31;
    if (e < N_EDGES) {
        int s = src[e], d = dst[e];
        float sc = cs[s];
        const float* hs = h   + (size_t)s * HIDDEN;
        float*       ad = agg + (size_t)d * HIDDEN;
#pragma unroll
        for (int f = lane; f < HIDDEN; f += 32)
            atomicAdd(&ad[f], hs[f] * sc);
    }
}

// ---------- WMMA GEMM: hout = relu((agg * cd) @ W + bias) ----------
// block = 256 threads = 8 waves; block covers rows [16*blockIdx, +16), each wave one 16-col tile.
__global__ void __launch_bounds__(256) gemm_relu_kernel(const float* __restrict__ agg,
                                                        const float* __restrict__ cd,
                                                        const float* __restrict__ W,
                                                        const float* __restrict__ bias,
                                                        float* __restrict__ hout) {
    __shared__ float Wl[HIDDEN * HIDDEN];   // 64 KB
    int tid = threadIdx.x;
    for (int i = tid; i < HIDDEN * HIDDEN; i += 256) Wl[i] = W[i];
    __syncthreads();

    int wave = tid >> 5;
    int lane = tid & 31;
    int half = lane >> 4;     // 0: K pair {0,1}; 1: K pair {2,3}
    int lr   = lane & 15;
    int m0   = blockIdx.x * 16;
    int m    = m0 + lr;               // A row held by this lane
    int n    = wave * 16 + lr;        // B/D column held by this lane

    float cdm = cd[m];
    const float* arow = agg + (size_t)m * HIDDEN + half * 2;
    const float* brow = Wl + (half * 2) * HIDDEN + n;

    v8f acc = {};
#pragma unroll
    for (int k0 = 0; k0 < HIDDEN; k0 += 4) {
        v2f av, bv;
        av.x = arow[k0]     * cdm;               // A[m][k0 + 2*half]
        av.y = arow[k0 + 1] * cdm;               // A[m][k0 + 2*half + 1]
        bv.x = brow[k0 * HIDDEN];                // W[k0 + 2*half][n]
        bv.y = brow[k0 * HIDDEN + HIDDEN];       // W[k0 + 2*half + 1][n]
        acc = __builtin_amdgcn_wmma_f32_16x16x4_f32(
            /*neg_a=*/false, av, /*neg_b=*/false, bv,
            /*c_mod=*/(short)0, acc, /*reuse_a=*/false, /*reuse_b=*/false);
    }

    float bn = bias[n];
#pragma unroll
    for (int r = 0; r < 8; ++r) {
        int mm = m0 + half * 8 + r;              // D layout: VGPR r -> rows r / r+8
        hout[(size_t)mm * HIDDEN + n] = fmaxf(acc[r] + bn, 0.0f);
    }
}

// ---------- mean pooling: one wave32 per node ----------
__global__ void pool_kernel(const int* __restrict__ seg, const float* __restrict__ h,
                            float* __restrict__ hg, float* __restrict__ counts) {
    int gid  = blockIdx.x * blockDim.x + threadIdx.x;
    int node = gid >> 5;
    int lane = gid & 31;
    if (node < N_NODES) {
        int g = seg[node];
        const float* hn  = h  + (size_t)node * HIDDEN;
        float*       hgo = hg + (size_t)g * HIDDEN;
#pragma unroll
        for (int f = lane; f < HIDDEN; f += 32)
            atomicAdd(&hgo[f], hn[f]);
        if (lane == 0) atomicAdd(&counts[g], 1.0f);
    }
}

__global__ void pool_div_kernel(float* __restrict__ hg, const float* __restrict__ counts) {
    int i = blockIdx.x * blockDim.x + threadIdx.x;   // N_GRAPHS*HIDDEN
    if (i < N_GRAPHS * HIDDEN)
        hg[i] /= fmaxf(counts[i >> 7], 1.0f);
}

// ---------- classifier: logits[g][c] = sum_k |hg1-hg2| * Wc + bc ----------
__global__ void logits_kernel(const float* __restrict__ hg1, const float* __restrict__ hg2,
                              const float* __restrict__ Wc, const float* __restrict__ bc,
                              float* __restrict__ logits) {
    int i = blockIdx.x * blockDim.x + threadIdx.x;
    if (i < N_GRAPHS * N_CLASSES) {
        int g = i / N_CLASSES, c = i % N_CLASSES;
        float s = bc[c];
#pragma unroll 8
        for (int k = 0; k < HIDDEN; ++k)
            s += fabsf(hg1[g * HIDDEN + k] - hg2[g * HIDDEN + k]) * Wc[k * N_CLASSES + c];
        logits[i] = s;
    }
}

// ---------- host-side branch driver ----------
static void run_branch(const int* src, const int* dst, const int* seg,
                       const float* W1, const float* b1,
                       const float* W2, const float* b2,
                       const float* W3, const float* b3,
                       const float* W4, const float* b4,
                       float* hg_out, float* ws_f, hipStream_t stream) {
    float* outdeg = ws_f;
    float* indeg  = ws_f + 100000;
    float* cs     = ws_f + 200000;
    float* cd     = ws_f + 300000;
    float* agg1   = ws_f + 400000;
    float* counts = ws_f + 500000;                      // 128 floats
    float* h      = ws_f + 500224;                      // 16B-aligned
    float* agg    = h + (size_t)N_NODES * HIDDEN;

    // zero degree/agg1/counts region
    hipMemsetAsync(ws_f, 0, (size_t)500224 * sizeof(float), stream);

    const int T = 256;
    deg_kernel<<<(N_EDGES + T - 1) / T, T, 0, stream>>>(src, dst, outdeg, indeg);
    norm_kernel<<<(N_NODES + T - 1) / T, T, 0, stream>>>(outdeg, indeg, cs, cd);

    // layer 1 (scalar feature)
    scatter1_kernel<<<(N_EDGES + T - 1) / T, T, 0, stream>>>(src, dst, indeg, cs, agg1);
    layer1_kernel<<<(N_NODES * HIDDEN + T - 1) / T, T, 0, stream>>>(agg1, cd, W1, b1, h);

    // layers 2..4: scatter + WMMA GEMM
    const float* Ws[3] = {W2, W3, W4};
    const float* bs[3] = {b2, b3, b4};
    for (int l = 0; l < 3; ++l) {
        hipMemsetAsync(agg, 0, (size_t)N_NODES * HIDDEN * sizeof(float), stream);
        scatter_kernel<<<(N_EDGES * 32) / T, T, 0, stream>>>(src, dst, cs, h, agg);
        gemm_relu_kernel<<<N_NODES / 16, T, 0, stream>>>(agg, cd, Ws[l], bs[l], h);
    }

    // mean pooling into hg_out (d_out region)
    hipMemsetAsync(hg_out, 0, (size_t)N_GRAPHS * HIDDEN * sizeof(float), stream);
    pool_kernel<<<(N_NODES * 32 + T - 1) / T, T, 0, stream>>>(seg, h, hg_out, counts);
    pool_div_kernel<<<(N_GRAPHS * HIDDEN + T - 1) / T, T, 0, stream>>>(hg_out, counts);
}

extern "C" void kernel_launch(void* const* d_in, const int* in_sizes, int n_in,
                              void* d_out, int out_size, void* d_ws, size_t ws_size,
                              hipStream_t stream) {
    const int*   src1 = (const int*)d_in[0];
    const int*   dst1 = (const int*)d_in[1];
    const int*   seg1 = (const int*)d_in[2];
    const int*   src2 = (const int*)d_in[3];
    const int*   dst2 = (const int*)d_in[4];
    const int*   seg2 = (const int*)d_in[5];
    const float* W1   = (const float*)d_in[6];
    const float* b1   = (const float*)d_in[7];
    const float* W2   = (const float*)d_in[8];
    const float* b2   = (const float*)d_in[9];
    const float* W3   = (const float*)d_in[10];
    const float* b3   = (const float*)d_in[11];
    const float* W4   = (const float*)d_in[12];
    const float* b4   = (const float*)d_in[13];
    const float* Wc   = (const float*)d_in[14];
    const float* bc   = (const float*)d_in[15];

    float* out   = (float*)d_out;
    float* hg1   = out;                              // 128*128
    float* hg2   = out + N_GRAPHS * HIDDEN;          // 128*128
    float* logit = out + 2 * N_GRAPHS * HIDDEN;      // 128*10
    float* ws_f  = (float*)d_ws;

    run_branch(src1, dst1, seg1, W1, b1, W2, b2, W3, b3, W4, b4, hg1, ws_f, stream);
    run_branch(src2, dst2, seg2, W1, b1, W2, b2, W3, b3, W4, b4, hg2, ws_f, stream);

    logits_kernel<<<(N_GRAPHS * N_CLASSES + 255) / 256, 256, 0, stream>>>(hg1, hg2, Wc, bc, logit);
}